// VectorQuantizer_31430570672177
// MI455X (gfx1250) — compile-verified
//
#include <hip/hip_runtime.h>
#include <hip/hip_bf16.h>

typedef __attribute__((ext_vector_type(2))) float v2f;
typedef __attribute__((ext_vector_type(8))) float v8f;

#define KCODES   256
#define HH       62
#define WW       64
#define BBATCH   768
#define MVEC     786432          // BBATCH * (WW * 64 / 4)
#define QL_SIZE  3047424         // BBATCH * HH * WW
#define OFF_MSE  (QL_SIZE)
#define OFF_IDX  (QL_SIZE + 1)
#define OFF_RATE (OFF_IDX + MVEC)
#define OFF_PRIOR (OFF_RATE + 1)
#define OFF_PARAM (OFF_PRIOR + 1)

#define NEG_INV_LN2 (-1.4426950408889634f)   // log2_pmf = log_pmf / (-ln 2)
#define INV_LMBDA   (100.0f)                 // 1 / 0.01
#define MSE_COUNT   (3145728.0f)             // MVEC * 4 (padded element count)

__global__ void vq_init_ws(float* ws) {
    if (threadIdx.x < 2) ws[threadIdx.x] = 0.0f;
}

__global__ __launch_bounds__(256) void vq_wmma_kernel(
    const float* __restrict__ lat,     // [768, 62, 64] flattened (N*C merged)
    const float* __restrict__ cb,      // [256, 4]
    const float* __restrict__ logpmf,  // [256]
    float* __restrict__ out,
    float* __restrict__ ws)
{
    const int lane = threadIdx.x & 31;
    const int hi   = lane >> 4;        // half-wave select
    const int col  = lane & 15;        // N column / M row within half
    const int waveId = blockIdx.x * (blockDim.x >> 5) + (threadIdx.x >> 5);
    const int base0 = waveId * 32;     // this wave covers vectors base0 .. base0+31

    // ---- Load B tiles (codebook^T, 16 tiles of 4x16) + per-code bias ----
    // B layout (4x16 f32): lane L, vgpr v -> B[k = 2*hi + v][n = col] = cb[n][2*hi + v]
    // Full float4 row load per lane: no cross-lane traffic for ||c||^2.
    v2f   bt[16];
    float c2rb[16];
    const float4* cbv = (const float4*)cb;
#pragma unroll
    for (int t = 0; t < 16; ++t) {
        int n = t * 16 + col;
        float4 c4 = cbv[n];
        bt[t].x = hi ? c4.z : c4.x;
        bt[t].y = hi ? c4.w : c4.y;
        float c2 = c4.x * c4.x + c4.y * c4.y + c4.z * c4.z + c4.w * c4.w;
        c2rb[t] = c2 + logpmf[n] * (NEG_INV_LN2 * INV_LMBDA);
    }

    const int b = base0 >> 10;         // image index (0..767); constant over 32 vectors
    float mse_part = 0.0f, rate_part = 0.0f;

#pragma unroll
    for (int sub = 0; sub < 2; ++sub) {
        const int base = base0 + sub * 16;
        const int w    = (base >> 4) & 63;   // column, fixed across this M-tile
        const float* xp  = lat + (size_t)b * (HH * WW) + w;

        // ---- Load A tile (16 vectors x 4, f32), pre-scaled by -2 ----
        // A layout: lane L holds x[m = base+col][k = 2*hi, 2*hi+1]
        v2f av;
        {
            int h0  = col * 4 + 2 * hi;
            int h1  = h0 + 1;
            int hs0 = (h0 < HH) ? h0 : h0 - 2;   // padded rows repeat rows 60,61
            int hs1 = (h1 < HH) ? h1 : h1 - 2;
            av.x = -2.0f * xp[(size_t)hs0 * WW];
            av.y = -2.0f * xp[(size_t)hs1 * WW];
        }

        // ---- 16 WMMAs: D = (-2x)·c^T + (||c||^2 + rate_bias) ; running argmin ----
        float minv[8];
        int   mini[8];
#pragma unroll
        for (int r = 0; r < 8; ++r) { minv[r] = 3.4e38f; mini[r] = 0; }

#pragma unroll
        for (int t = 0; t < 16; ++t) {
            const float bias = c2rb[t];
            v8f acc;
#pragma unroll
            for (int i = 0; i < 8; ++i) acc[i] = bias;   // C = bias (same for all rows of this column)
            acc = __builtin_amdgcn_wmma_f32_16x16x4_f32(
                      false, av, false, bt[t], (short)0, acc, false, false);
            const int nidx = t * 16 + col;
#pragma unroll
            for (int r = 0; r < 8; ++r) {
                float s = acc[r];
                if (s < minv[r]) { minv[r] = s; mini[r] = nidx; }
            }
        }

        // ---- Butterfly min-reduce over the 16 code-columns (within each half) ----
#pragma unroll
        for (int off = 1; off < 16; off <<= 1) {
#pragma unroll
            for (int r = 0; r < 8; ++r) {
                float ov = __shfl_xor(minv[r], off, 32);
                int   oi = __shfl_xor(mini[r], off, 32);
                if (ov < minv[r] || (ov == minv[r] && oi < mini[r])) {
                    minv[r] = ov; mini[r] = oi;
                }
            }
        }

        // ---- Writers: lane 0 handles rows 0..7, lane 16 handles rows 8..15 ----
        if (col == 0) {
            float* qlp = out + (size_t)b * (HH * WW) + w;
#pragma unroll
            for (int r = 0; r < 8; ++r) {
                int m   = base + r + 8 * hi;
                int idx = mini[r];
                out[OFF_IDX + m] = (float)idx;
                float4 q4 = cbv[idx];
                float  qd[4] = {q4.x, q4.y, q4.z, q4.w};
                int cc = r + 8 * hi;               // chunk index of vector m
#pragma unroll
                for (int d = 0; d < 4; ++d) {
                    int h  = cc * 4 + d;
                    int hs = (h < HH) ? h : h - 2;
                    float x  = xp[(size_t)hs * WW];
                    float df = qd[d] - x;
                    mse_part += df * df;
                    if (h < HH) qlp[(size_t)h * WW] = qd[d];   // crop padded rows
                }
                rate_part += logpmf[idx] * NEG_INV_LN2;
            }
        }
    }

    // combine the two writer lanes, one atomic pair per wave
    mse_part  += __shfl_xor(mse_part, 16, 32);
    rate_part += __shfl_xor(rate_part, 16, 32);
    if (lane == 0) {
        atomicAdd(&ws[0], mse_part);
        atomicAdd(&ws[1], rate_part);
    }
}

__global__ void vq_finalize(const float* __restrict__ ws, float* __restrict__ out) {
    if (threadIdx.x == 0) {
        out[OFF_MSE]   = ws[0] / MSE_COUNT;
        out[OFF_RATE]  = ws[1];
        out[OFF_PRIOR] = 0.0f;
        out[OFF_PARAM] = 0.0f;
    }
}

extern "C" void kernel_launch(void* const* d_in, const int* in_sizes, int n_in,
                              void* d_out, int out_size, void* d_ws, size_t ws_size,
                              hipStream_t stream) {
    const float* lat    = (const float*)d_in[0];   // latents  [4,192,62,64]
    const float* cb     = (const float*)d_in[1];   // codebook [256,4]
    const float* logpmf = (const float*)d_in[2];   // log_pmf  [256]
    float* out = (float*)d_out;
    float* ws  = (float*)d_ws;

    vq_init_ws<<<1, 32, 0, stream>>>(ws);

    // one wave handles 32 vectors (2 WMMA M-tiles); 8 waves (256 threads) per block
    const int blocks = MVEC / (32 * 8);            // 3072
    vq_wmma_kernel<<<blocks, 256, 0, stream>>>(lat, cb, logpmf, out, ws);

    vq_finalize<<<1, 32, 0, stream>>>(ws, out);
}